// MultiScaleRoIAlign_81243601371176
// MI455X (gfx1250) — compile-verified
//
#include <hip/hip_runtime.h>
#include <stdint.h>

// ---------------------------------------------------------------------------
// MultiScaleRoIAlign for MI455X (gfx1250).
// Memory-latency bound gather problem: use the CDNA5 Tensor Data Mover to DMA
// each ROI's rectangular feature sub-tile (x-contiguous, W-strided rows,
// H*W-strided channels) into LDS, then do the 16 bilinear taps from LDS.
// Fallback to direct (L2-resident) gathers for rare oversized tiles.
// ---------------------------------------------------------------------------

typedef unsigned int uint32;
typedef uint32 __attribute__((ext_vector_type(4))) v4u;
typedef int    __attribute__((ext_vector_type(4))) v4i;
typedef int    __attribute__((ext_vector_type(8))) v8i;

#define CG        64     // channels per group (256 / gridDim.y)
#define TILE_PIX  224    // max tile pixels staged in LDS (57 KB for 64 ch)

__global__ __launch_bounds__(256) void msroi_kernel(
    const float* __restrict__ f0, const float* __restrict__ f1,
    const float* __restrict__ f2, const float* __restrict__ f3,
    const float* __restrict__ boxes, float* __restrict__ out)
{
    __shared__ float tile[CG * TILE_PIX];
    __shared__ int   s_yl[14], s_yh[14], s_xl[14], s_xh[14];
    __shared__ float s_wyl[14], s_wyh[14], s_vy[14];
    __shared__ float s_wxl[14], s_wxh[14], s_vx[14];

    const int r  = blockIdx.x;   // ROI index (B*256)
    const int cg = blockIdx.y;   // channel group
    const int t  = threadIdx.x;
    const int b  = r >> 8;       // batch (Nb = 256)

    const float bx1 = boxes[r * 4 + 0];
    const float by1 = boxes[r * 4 + 1];
    const float bx2 = boxes[r * 4 + 2];
    const float by2 = boxes[r * 4 + 3];

    // ---- FPN level selection (mirror reference) ----
    const float area  = (bx2 - bx1) * (by2 - by1);
    const float scale = sqrtf(fmaxf(area, 0.0f));
    float lv = floorf(4.0f + log2f(scale / 224.0f + 1e-6f));
    lv = fminf(fmaxf(lv, 2.0f), 5.0f);
    const int li = (int)lv - 2;

    int H, W;
    const float* fb;
    switch (li) {
        case 0:  H = 200; W = 200; fb = f0; break;
        case 1:  H = 100; W = 100; fb = f1; break;
        case 2:  H = 50;  W = 50;  fb = f2; break;
        default: H = 25;  W = 25;  fb = f3; break;
    }
    const int   HW  = H * W;
    const float inv = 1.0f / (float)(4 << li);

    const float x1 = bx1 * inv - 0.5f, y1f = by1 * inv - 0.5f;
    const float x2 = bx2 * inv - 0.5f, y2f = by2 * inv - 0.5f;
    const float bh = (y2f - y1f) * (1.0f / 7.0f);
    const float bw = (x2  - x1 ) * (1.0f / 7.0f);

    // ---- axis terms: 14 sample coords per axis ----
    if (t < 14) {
        const int k = t;
        const float g  = y1f + ((float)(k >> 1) + ((k & 1) ? 0.75f : 0.25f)) * bh;
        const float vd = (g >= -1.0f && g <= (float)H) ? 1.0f : 0.0f;
        const float gc = fminf(fmaxf(g, 0.0f), (float)(H - 1));
        const int   lo = (int)floorf(gc);
        const int   hi = min(lo + 1, H - 1);
        const float l  = gc - (float)lo;
        s_yl[k] = lo; s_yh[k] = hi;
        s_wyl[k] = 1.0f - l; s_wyh[k] = l; s_vy[k] = vd;
    } else if (t >= 16 && t < 30) {
        const int k = t - 16;
        const float g  = x1 + ((float)(k >> 1) + ((k & 1) ? 0.75f : 0.25f)) * bw;
        const float vd = (g >= -1.0f && g <= (float)W) ? 1.0f : 0.0f;
        const float gc = fminf(fmaxf(g, 0.0f), (float)(W - 1));
        const int   lo = (int)floorf(gc);
        const int   hi = min(lo + 1, W - 1);
        const float l  = gc - (float)lo;
        s_xl[k] = lo; s_xh[k] = hi;
        s_wxl[k] = 1.0f - l; s_wxh[k] = l; s_vx[k] = vd;
    }
    __syncthreads();

    // ---- tile bounds (uniform; cheap redundant compute) ----
    int y0 = s_yl[0], ym = s_yh[0], x0 = s_xl[0], xm = s_xh[0];
    #pragma unroll
    for (int k = 1; k < 14; ++k) {
        y0 = min(y0, s_yl[k]); ym = max(ym, s_yh[k]);
        x0 = min(x0, s_xl[k]); xm = max(xm, s_xh[k]);
    }
    const int  ycnt = ym - y0 + 1;
    const int  xcnt = xm - x0 + 1;
    const int  pix  = ycnt * xcnt;
    const bool useTile = (pix <= TILE_PIX);

    const float* base = fb + ((size_t)b * 256 + (size_t)cg * CG) * (size_t)HW;

    // ---- TDM: DMA [CG][ycnt][xcnt] sub-tile into LDS (wave 0 issues) ----
    if (useTile && t < 32) {
        const uint64_t ga   = (uint64_t)(uintptr_t)(base + (size_t)y0 * W + x0);
        const uint32 lds    = (uint32)(uintptr_t)&tile[0];       // LDS byte offset
        const int Wu  = __builtin_amdgcn_readfirstlane(W);
        const int HWu = __builtin_amdgcn_readfirstlane(HW);
        const int xcu = __builtin_amdgcn_readfirstlane(xcnt);
        const int ycu = __builtin_amdgcn_readfirstlane(ycnt);
        const int Hu  = __builtin_amdgcn_readfirstlane(H);
        const uint32 galo = (uint32)__builtin_amdgcn_readfirstlane((int)(uint32)ga);
        const uint32 gahi = (uint32)__builtin_amdgcn_readfirstlane((int)(uint32)(ga >> 32));
        const uint32 ldsu = (uint32)__builtin_amdgcn_readfirstlane((int)lds);

        // D# group 0: count=1 | lds_addr | global_addr[56:0] | type=2
        v4u g0 = { 1u, ldsu, galo, (gahi & 0x1FFFFFFu) | (2u << 30) };

        // D# group 1: data_size=4B; tensor dims W,H; tile dims xcnt,ycnt,CG;
        //             dim0_stride=W, dim1_stride=H*W (channel plane stride)
        v8i g1 = {
            (int)(2u << 16),                                     // data_size=4B
            (int)(((uint32)Wu  & 0xFFFFu) << 16),                // tensor_dim0 lo
            (int)(((uint32)Hu  & 0xFFFFu) << 16),                // dim0 hi=0 | tensor_dim1 lo
            (int)(((uint32)xcu & 0xFFFFu) << 16),                // dim1 hi=0 | tile_dim0
            (int)(((uint32)ycu & 0xFFFFu) | ((uint32)CG << 16)), // tile_dim1 | tile_dim2
            (int)(uint32)Wu,                                     // tensor_dim0_stride lo32
            (int)(((uint32)HWu & 0xFFFFu) << 16),                // stride0 hi=0 | stride1 lo16
            (int)(((uint32)HWu >> 16) & 0xFFFFu)                 // tensor_dim1_stride hi
        };

        // D# group 2: tensor_dim2 = CG channels; 4th dim unused
        v4i g2 = { CG, 0, 0, 0 };
        v4i g3 = { 0, 0, 0, 0 };
        // Extra descriptor words required by the 6-arg toolchain variant
        // (reserved for non-iterating / non-gather 3D loads).
        v8i g4 = { 0, 0, 0, 0, 0, 0, 0, 0 };

        __builtin_amdgcn_tensor_load_to_lds(g0, g1, g2, g3, g4, 0);
        __builtin_amdgcn_s_wait_tensorcnt(0);
    }
    __syncthreads();

    // ---- bilinear gather: thread = (channel, bins strided by 4) ----
    const int cl    = t & (CG - 1);
    const int squad = t >> 6;
    const float* cbase = base + (size_t)cl * HW;
    const float* tp    = &tile[cl * pix];
    float* obase = out + ((size_t)r * 256 + (size_t)cg * CG + cl) * 49;

    for (int bin = squad; bin < 49; bin += 4) {
        const int byi = bin / 7;
        const int bxi = bin - byi * 7;
        float acc = 0.0f;
        #pragma unroll
        for (int dy = 0; dy < 2; ++dy) {
            const int   sy  = 2 * byi + dy;
            const int   yl  = s_yl[sy], yh = s_yh[sy];
            const float wyl = s_wyl[sy], wyh = s_wyh[sy], vy = s_vy[sy];
            #pragma unroll
            for (int dx = 0; dx < 2; ++dx) {
                const int   sx  = 2 * bxi + dx;
                const int   xl  = s_xl[sx], xh = s_xh[sx];
                const float wxl = s_wxl[sx], wxh = s_wxh[sx], vx = s_vx[sx];
                float v00, v01, v10, v11;
                if (useTile) {
                    const int r0 = (yl - y0) * xcnt, r1 = (yh - y0) * xcnt;
                    v00 = tp[r0 + (xl - x0)];
                    v01 = tp[r0 + (xh - x0)];
                    v10 = tp[r1 + (xl - x0)];
                    v11 = tp[r1 + (xh - x0)];
                } else {
                    v00 = cbase[yl * W + xl];
                    v01 = cbase[yl * W + xh];
                    v10 = cbase[yh * W + xl];
                    v11 = cbase[yh * W + xh];
                }
                const float s = wyl * (wxl * v00 + wxh * v01)
                              + wyh * (wxl * v10 + wxh * v11);
                acc += s * (vy * vx);
            }
        }
        obase[bin] = acc * 0.25f;
    }
}

extern "C" void kernel_launch(void* const* d_in, const int* in_sizes, int n_in,
                              void* d_out, int out_size, void* d_ws, size_t ws_size,
                              hipStream_t stream) {
    const float* f0    = (const float*)d_in[0];
    const float* f1    = (const float*)d_in[1];
    const float* f2    = (const float*)d_in[2];
    const float* f3    = (const float*)d_in[3];
    const float* boxes = (const float*)d_in[4];
    float* out = (float*)d_out;

    const int n_rois = in_sizes[4] / 4;          // B * 256
    dim3 grid(n_rois, 256 / CG, 1);
    msroi_kernel<<<grid, 256, 0, stream>>>(f0, f1, f2, f3, boxes, out);
}